// GATModule_10359461118096
// MI455X (gfx1250) — compile-verified
//
#include <hip/hip_runtime.h>

// GAT forward for MI455X (gfx1250), wave32.
// Pipeline: LN -> WMMA GEMM (W_in) -> scores -> edge max/softmax (atomics)
//        -> weighted aggregation (atomics) -> WMMA GEMM (W_ff) -> d_out.
//
// Roofline: GEMMs are ~3.3 GFLOP total (negligible); the edge phase
// (~410 MB of L2-resident gathers + ~10^8 f32 atomics) dominates. So:
//  - GEMMs use full-precision V_WMMA_F32_16X16X4_F32 (matrix pipe, exact f32)
//    with the 64KB weight matrix staged to LDS via async-to-LDS DMA.
//  - Edge phase uses native global_atomic_add_f32 (unsafeAtomicAdd) and
//    integer atomic max with a monotone float<->uint encoding.

#define NNODES 50000
#define NEDGES 800000
#define DIMF   128
#define NHEADS 8
#define LN_EPS 1e-5f
#define NEG_SLOPE 0.2f

typedef __attribute__((ext_vector_type(2))) float v2f;
typedef __attribute__((ext_vector_type(8))) float v8f;

__device__ __forceinline__ float wave_sum(float v) {
#pragma unroll
  for (int off = 16; off > 0; off >>= 1) v += __shfl_xor(v, off, 32);
  return v;
}

// Native f32 atomic add (global_atomic_add_f32, no CAS loop).
__device__ __forceinline__ void atomic_add_f32(float* p, float v) {
  (void)unsafeAtomicAdd(p, v);
}

// Monotone order-preserving float<->uint encoding for atomic max on floats.
__device__ __forceinline__ unsigned enc_f32(float f) {
  unsigned b = __float_as_uint(f);
  return (b & 0x80000000u) ? ~b : (b | 0x80000000u);
}
__device__ __forceinline__ float dec_f32(unsigned u) {
  return __uint_as_float((u & 0x80000000u) ? (u ^ 0x80000000u) : ~u);
}

// ---------------------------------------------------------------- zero
__global__ void zero_kernel(float* __restrict__ p, long n) {
  long i = (long)blockIdx.x * blockDim.x + threadIdx.x;
  long stride = (long)gridDim.x * blockDim.x;
  for (; i < n; i += stride) p[i] = 0.0f;
}

// ---------------------------------------------------------------- LayerNorm
// One wave per node; each lane owns 4 contiguous features (float4).
__global__ void ln_kernel(const float* __restrict__ x,
                          const float* __restrict__ gamma,
                          const float* __restrict__ beta,
                          float* __restrict__ xln) {
  int wid  = (int)((blockIdx.x * blockDim.x + threadIdx.x) >> 5);
  int lane = threadIdx.x & 31;
  if (wid >= NNODES) return;
  const float4 v = *(const float4*)(x + (size_t)wid * DIMF + lane * 4);
  float s  = v.x + v.y + v.z + v.w;
  float sq = v.x * v.x + v.y * v.y + v.z * v.z + v.w * v.w;
  s  = wave_sum(s);
  sq = wave_sum(sq);
  float mu  = s * (1.0f / DIMF);
  float var = sq * (1.0f / DIMF) - mu * mu;
  float rs  = rsqrtf(var + LN_EPS);
  float4 g = *(const float4*)(gamma + lane * 4);
  float4 b = *(const float4*)(beta + lane * 4);
  float4 o;
  o.x = (v.x - mu) * rs * g.x + b.x;
  o.y = (v.y - mu) * rs * g.y + b.y;
  o.z = (v.z - mu) * rs * g.z + b.z;
  o.w = (v.w - mu) * rs * g.w + b.w;
  *(float4*)(xln + (size_t)wid * DIMF + lane * 4) = o;
}

// ---------------------------------------------------------------- WMMA GEMM
// out[M,128] = A[M,128] @ W[128,128] + bias.  One wave -> 16x128 strip.
// The 64KB W matrix is staged into LDS once per block with
// GLOBAL_LOAD_ASYNC_TO_LDS_B128 (ASYNCcnt-tracked DMA), then each wave's
// B fragments come from ds_load while A streams from global.
// A 16x4 f32 layout (ISA 7.12.2): lanes 0-15: vgpr0=K0, vgpr1=K1;
//                                 lanes 16-31: vgpr0=K2, vgpr1=K3.
// C/D: vgpr r -> (M=r, N=lane) for lanes 0-15, (M=r+8, N=lane-16) otherwise.
__global__ void gemm_wmma_f32(const float* __restrict__ A,
                              const float* __restrict__ W,
                              const float* __restrict__ bias,
                              float* __restrict__ out, int M) {
  __shared__ float Wlds[DIMF * DIMF];  // 64KB (of 320KB per WGP)

  // ---- cooperative async stage of W into LDS (all 128 threads) ----
  {
    int row = threadIdx.x;  // 0..127 -> one 128-float row per thread
    unsigned ldsbase = (unsigned)(size_t)(void*)Wlds;
#pragma unroll
    for (int j = 0; j < 8; ++j) {
      unsigned loff = ldsbase + (unsigned)(row * DIMF + j * 16) * 4u;
      const float* gp = W + (size_t)row * DIMF + j * 16;
      asm volatile("global_load_async_to_lds_b128 %0, %1, off"
                   :: "v"(loff), "v"(gp)
                   : "memory");
    }
    asm volatile("s_wait_asynccnt 0" ::: "memory");
  }
  __syncthreads();

  int wave = (int)((blockIdx.x * blockDim.x + threadIdx.x) >> 5);
  int lane = threadIdx.x & 31;
  int m0 = wave * 16;
  if (m0 >= M) return;            // wave-uniform: EXEC stays all-1s for WMMA
  int half = lane >> 4;           // 0 for lanes 0-15, 1 for 16-31
  int l16  = lane & 15;

  v8f acc[8];
#pragma unroll
  for (int nt = 0; nt < 8; ++nt)
#pragma unroll
    for (int r = 0; r < 8; ++r) acc[nt][r] = 0.0f;

  const float* arow = A + (size_t)(m0 + l16) * DIMF;
  for (int k = 0; k < DIMF; k += 4) {
    int kk = k + 2 * half;
    v2f a;
    a.x = arow[kk];
    a.y = arow[kk + 1];
#pragma unroll
    for (int nt = 0; nt < 8; ++nt) {
      int n0 = nt * 16;
      v2f b;
      b.x = Wlds[kk * DIMF + n0 + l16];
      b.y = Wlds[(kk + 1) * DIMF + n0 + l16];
      acc[nt] = __builtin_amdgcn_wmma_f32_16x16x4_f32(
          false, a, false, b, (short)0, acc[nt], false, false);
    }
  }

#pragma unroll
  for (int nt = 0; nt < 8; ++nt) {
    int n0 = nt * 16;
    float bv = bias[n0 + l16];
#pragma unroll
    for (int r = 0; r < 8; ++r) {
      int m = m0 + r + 8 * half;
      out[(size_t)m * DIMF + n0 + l16] = acc[nt][r] + bv;
    }
  }
}

// ---------------------------------------------------------------- scores
// su = y@W_u + b_u,  sv = y@W_v   ([N,8] each). One wave per node.
__global__ void scores_kernel(const float* __restrict__ y,
                              const float* __restrict__ Wu,
                              const float* __restrict__ bu,
                              const float* __restrict__ Wv,
                              float* __restrict__ su, float* __restrict__ sv) {
  int wid  = (int)((blockIdx.x * blockDim.x + threadIdx.x) >> 5);
  int lane = threadIdx.x & 31;
  if (wid >= NNODES) return;
  float4 yv = *(const float4*)(y + (size_t)wid * DIMF + lane * 4);
  float ya[4] = {yv.x, yv.y, yv.z, yv.w};
  float pu[NHEADS], pv[NHEADS];
#pragma unroll
  for (int h = 0; h < NHEADS; ++h) { pu[h] = 0.0f; pv[h] = 0.0f; }
#pragma unroll
  for (int j = 0; j < 4; ++j) {
    int k = lane * 4 + j;
    const float* wuk = Wu + (size_t)k * NHEADS;
    const float* wvk = Wv + (size_t)k * NHEADS;
#pragma unroll
    for (int h = 0; h < NHEADS; ++h) {
      pu[h] += ya[j] * wuk[h];
      pv[h] += ya[j] * wvk[h];
    }
  }
#pragma unroll
  for (int h = 0; h < NHEADS; ++h) {
    float ru = wave_sum(pu[h]);
    float rv = wave_sum(pv[h]);
    if (lane == 0) {
      su[(size_t)wid * NHEADS + h] = ru + bu[h];
      sv[(size_t)wid * NHEADS + h] = rv;
    }
  }
}

// ------------------------------------------------- edge pass 1: leaky + max
// one thread per (edge, head); 8 consecutive threads share the edge -> the
// su/sv/menc rows they touch are one 32B run (head-contiguous layout).
__global__ void edge_max_kernel(const int* __restrict__ src,
                                const int* __restrict__ dst,
                                const float* __restrict__ su,
                                const float* __restrict__ sv,
                                float* __restrict__ ebuf,
                                unsigned* __restrict__ menc) {
  int t = blockIdx.x * blockDim.x + threadIdx.x;
  if (t >= NEDGES * NHEADS) return;
  int e = t >> 3, h = t & 7;
  int sn = src[e], dn = dst[e];
  float v = su[(size_t)sn * NHEADS + h] + sv[(size_t)dn * NHEADS + h];
  v = (v >= 0.0f) ? v : NEG_SLOPE * v;
  ebuf[t] = v;
  __hip_atomic_fetch_max(&menc[(size_t)dn * NHEADS + h], enc_f32(v),
                         __ATOMIC_RELAXED, __HIP_MEMORY_SCOPE_AGENT);
}

// ------------------------------------------------- edge pass 2: exp + sum
__global__ void edge_exp_kernel(const int* __restrict__ dst,
                                const unsigned* __restrict__ menc,
                                float* __restrict__ ebuf,
                                float* __restrict__ sden) {
  int t = blockIdx.x * blockDim.x + threadIdx.x;
  if (t >= NEDGES * NHEADS) return;
  int e = t >> 3, h = t & 7;
  int dn = dst[e];
  float m  = dec_f32(menc[(size_t)dn * NHEADS + h]);
  float ex = __expf(ebuf[t] - m);
  ebuf[t] = ex;
  atomic_add_f32(&sden[(size_t)dn * NHEADS + h], ex);
}

// Precompute 1/s once per (node, head) instead of per-edge divides.
__global__ void recip_kernel(float* __restrict__ s, int n) {
  int i = blockIdx.x * blockDim.x + threadIdx.x;
  if (i >= n) return;
  float v = s[i];
  s[i] = (v > 0.0f) ? (1.0f / v) : 0.0f;  // empty segments never read
}

// ------------------------------------------------- edge pass 3: aggregation
// feature d of a node belongs to head (d & 7) per the [N, hd, H] reshape.
// 32 threads per edge, each handles a float4 chunk -> fully coalesced y row,
// head index within a chunk is (4c & 7)..+3 so probs load as one float4.
__global__ void edge_agg_kernel(const int* __restrict__ src,
                                const int* __restrict__ dst,
                                const float* __restrict__ y,
                                const float* __restrict__ ex,
                                const float* __restrict__ rs,
                                float* __restrict__ agg) {
  int t = blockIdx.x * blockDim.x + threadIdx.x;
  if (t >= NEDGES * 32) return;
  int e = t >> 5, c = t & 31;
  int sn = src[e], dn = dst[e];
  int d0 = c * 4;
  int h0 = d0 & 7;  // 0 or 4
  float4 ex4 = *(const float4*)(ex + (size_t)e  * NHEADS + h0);
  float4 rs4 = *(const float4*)(rs + (size_t)dn * NHEADS + h0);
  float4 y4  = *(const float4*)(y  + (size_t)sn * DIMF   + d0);
  float* ag = agg + (size_t)dn * DIMF + d0;
  atomic_add_f32(ag + 0, y4.x * (ex4.x * rs4.x));
  atomic_add_f32(ag + 1, y4.y * (ex4.y * rs4.y));
  atomic_add_f32(ag + 2, y4.z * (ex4.z * rs4.z));
  atomic_add_f32(ag + 3, y4.w * (ex4.w * rs4.w));
}

// ---------------------------------------------------------------- launch
extern "C" void kernel_launch(void* const* d_in, const int* in_sizes, int n_in,
                              void* d_out, int out_size, void* d_ws, size_t ws_size,
                              hipStream_t stream) {
  (void)in_sizes; (void)n_in; (void)out_size; (void)ws_size;
  const float* x     = (const float*)d_in[0];
  const int*   src   = (const int*)d_in[1];
  const int*   dst   = (const int*)d_in[2];
  const float* gamma = (const float*)d_in[3];
  const float* beta  = (const float*)d_in[4];
  const float* W_in  = (const float*)d_in[5];
  const float* b_in  = (const float*)d_in[6];
  const float* W_u   = (const float*)d_in[7];
  const float* b_u   = (const float*)d_in[8];
  const float* W_v   = (const float*)d_in[9];
  const float* W_ff  = (const float*)d_in[10];
  const float* b_ff  = (const float*)d_in[11];
  float* out = (float*)d_out;

  // Workspace layout (floats). Total 27.2M floats = 108.8 MB (L2-resident).
  float* ws   = (float*)d_ws;
  float*    xln  = ws;                         //  6.4M  LN output
  float*    y    = ws + 6400000;               //  6.4M  x@W_in+b_in
  float*    su   = ws + 12800000;              //  0.4M  [N,8]
  float*    sv   = ws + 13200000;              //  0.4M  [N,8]
  float*    ebuf = ws + 13600000;              //  6.4M  [E,8] e -> ex
  unsigned* menc = (unsigned*)(ws + 20000000); //  0.4M  encoded seg-max
  float*    sden = ws + 20400000;              //  0.4M  seg-sum -> 1/s
  float*    agg  = ws + 20800000;              //  6.4M  aggregation

  // Zero the accumulation region (menc|sden|agg are contiguous: 7.2M elems).
  zero_kernel<<<2048, 256, 0, stream>>>(ws + 20000000, 7200000L);

  ln_kernel<<<(NNODES + 7) / 8, 256, 0, stream>>>(x, gamma, beta, xln);

  // 3125 strips of 16 rows; 4 waves (128 thr) per block.
  gemm_wmma_f32<<<(3125 + 3) / 4, 128, 0, stream>>>(xln, W_in, b_in, y, NNODES);

  scores_kernel<<<(NNODES + 7) / 8, 256, 0, stream>>>(y, W_u, b_u, W_v, su, sv);

  edge_max_kernel<<<(NEDGES * NHEADS + 255) / 256, 256, 0, stream>>>(
      src, dst, su, sv, ebuf, menc);
  edge_exp_kernel<<<(NEDGES * NHEADS + 255) / 256, 256, 0, stream>>>(
      dst, menc, ebuf, sden);
  recip_kernel<<<(NNODES * NHEADS + 255) / 256, 256, 0, stream>>>(
      sden, NNODES * NHEADS);
  edge_agg_kernel<<<(NEDGES * 32 + 255) / 256, 256, 0, stream>>>(
      src, dst, y, ebuf, sden, agg);

  gemm_wmma_f32<<<(3125 + 3) / 4, 128, 0, stream>>>(agg, W_ff, b_ff, out, NNODES);
}